// _CwECELoss_56461640073709
// MI455X (gfx1250) — compile-verified
//
#include <hip/hip_runtime.h>
#include <hip/hip_bf16.h>
#include <stdint.h>

// Problem shape (from reference): logits [65536, 1000] f32, labels [65536] i32,
// 15 confidence bins, avg over classes.
#define NB 15
#define NCLS 1000
#define HBINS (NB * NCLS)   // 15000 per-(bin,class) cells
#define CHUNKS 250          // 250 x 16B = 4000B = one 1000-float row, exact
#define HIST_BLOCKS 512
#define WPB 8               // waves per block (256 threads, wave32)

typedef float v2f __attribute__((ext_vector_type(2)));
typedef float v8f __attribute__((ext_vector_type(8)));

// Generic LDS pointer -> 32-bit LDS byte offset (flat LDS aperture keeps the
// offset in addr[31:0]).
__device__ __forceinline__ unsigned lds_off(const void* p) {
  return (unsigned)(unsigned long long)p;
}

// Issue exactly 8 async 16B/lane global->LDS copies per wave (one full row).
// ASYNCcnt += 8 per call; completion is in-order, so s_wait_asynccnt 8 after
// issuing the *next* row's 8 guarantees the *current* row is resident.
__device__ __forceinline__ void async_row_load(const float* __restrict__ grow,
                                               float* lbuf, int lane) {
  const char* gb = (const char*)grow;
#pragma unroll
  for (int i = 0; i < 8; ++i) {
    const int c = i * 32 + lane;
    if (c < CHUNKS) {  // iteration 7: lanes 0..25 active -> still one issue
      unsigned loff = lds_off(&lbuf[c * 4]);
      const void* ga = gb + (size_t)c * 16;
      asm volatile("global_load_async_to_lds_b128 %0, %1, off"
                   :: "v"(loff), "v"(ga)
                   : "memory");
    }
  }
}

__global__ void zero_f32(float* __restrict__ p, int n) {
  int i = blockIdx.x * 256 + threadIdx.x;
  if (i < n) p[i] = 0.0f;
}

__global__ __launch_bounds__(256, 1) void cwece_hist(
    const float* __restrict__ logits, const int* __restrict__ labels,
    float* __restrict__ conf_g, float* __restrict__ corr_g, int nrows) {
  __shared__ float sconf[HBINS];          // 60 KB block-private conf histogram
  __shared__ float rows[WPB][2][1024];    // 64 KB double-buffered row staging

  const int tid = threadIdx.x;
  for (int i = tid; i < HBINS; i += 256) sconf[i] = 0.0f;
  __syncthreads();

  const int wave = tid >> 5;
  const int lane = tid & 31;
  const int gw   = blockIdx.x * WPB + wave;
  const int nw   = gridDim.x * WPB;

  int row = gw;
  int sel = 0;
  if (row < nrows)
    async_row_load(logits + (size_t)row * NCLS, rows[wave][0], lane);

  for (; row < nrows; row += nw) {
    const int nxt = row + nw;
    if (nxt < nrows) {
      // prefetch next row into the other buffer, then wait for current (<=8
      // outstanding means the first 8 — this row — are done).
      async_row_load(logits + (size_t)nxt * NCLS, rows[wave][sel ^ 1], lane);
      asm volatile("s_wait_asynccnt 8" ::: "memory");
    } else {
      asm volatile("s_wait_asynccnt 0" ::: "memory");
    }
    float* buf = rows[wave][sel];

    // pass 1: row max (wave reduction)
    float m = -3.402823466e38f;
    for (int j = lane; j < NCLS; j += 32) m = fmaxf(m, buf[j]);
    for (int off = 16; off > 0; off >>= 1)
      m = fmaxf(m, __shfl_xor(m, off, 32));

    // pass 2: exp in place + sum (wave reduction)
    float s = 0.0f;
    for (int j = lane; j < NCLS; j += 32) {
      float e = __expf(buf[j] - m);
      buf[j] = e;
      s += e;
    }
    for (int off = 16; off > 0; off >>= 1)
      s += __shfl_xor(s, off, 32);
    const float inv = 1.0f / s;

    const int lbl = labels[row];

    // pass 3: bin and accumulate. bin = clip(ceil(15p)-1, 0, 14); p==0 invalid.
    for (int j = lane; j < NCLS; j += 32) {
      const float p = buf[j] * inv;
      if (p > 0.0f) {
        int b = (int)ceilf(p * 15.0f) - 1;
        b = b < 0 ? 0 : (b > NB - 1 ? NB - 1 : b);
        atomicAdd(&sconf[b * NCLS + j], p);        // ds_add_f32
        if (j == lbl) atomicAdd(&corr_g[b * NCLS + j], 1.0f);  // 1 per row
      }
    }
    sel ^= 1;
  }

  __syncthreads();
  for (int i = tid; i < HBINS; i += 256) {
    const float v = sconf[i];
    if (v != 0.0f) atomicAdd(&conf_g[i], v);
  }
}

// Final reduction: sum |conf - correct| over 15000 cells, scale by 1/(N*C).
// Per-wave 32->1 fold done with v_wmma_f32_16x16x4_f32 against a ones B:
// rowsum[m] = partial[m] + partial[m+16]; D vgprs 0..7 give rows 0..7 (lanes
// 0-15) / 8..15 (lanes 16-31); one shfl_xor(16) completes the wave sum.
__global__ __launch_bounds__(256) void cwece_finalize(
    const float* __restrict__ conf_g, const float* __restrict__ corr_g,
    float* __restrict__ out, int nrows) {
  float p = 0.0f;
  for (int i = threadIdx.x; i < HBINS; i += 256)
    p += fabsf(conf_g[i] - corr_g[i]);

  v2f a; a[0] = p;    a[1] = 0.0f;   // A[m,k]: k0 from lanes 0-15, k2 from 16-31
  v2f b; b[0] = 1.0f; b[1] = 1.0f;   // ones matrix
  v8f c = {};
  c = __builtin_amdgcn_wmma_f32_16x16x4_f32(
      /*neg_a=*/false, a, /*neg_b=*/false, b,
      /*c_mod=*/(short)0, c, /*reuse_a=*/false, /*reuse_b=*/false);
  float t = c[0] + c[1] + c[2] + c[3] + c[4] + c[5] + c[6] + c[7];
  t += __shfl_xor(t, 16, 32);        // all lanes now hold the wave total

  __shared__ float wsum[WPB];
  const int wave = threadIdx.x >> 5;
  const int lane = threadIdx.x & 31;
  if (lane == 0) wsum[wave] = t;
  __syncthreads();
  if (threadIdx.x == 0) {
    float stot = 0.0f;
    for (int w = 0; w < WPB; ++w) stot += wsum[w];
    out[0] = stot / ((float)nrows * (float)NCLS);
  }
}

extern "C" void kernel_launch(void* const* d_in, const int* in_sizes, int n_in,
                              void* d_out, int out_size, void* d_ws, size_t ws_size,
                              hipStream_t stream) {
  (void)n_in; (void)out_size; (void)ws_size;
  const float* logits = (const float*)d_in[0];
  const int*   labels = (const int*)d_in[1];
  const int    nrows  = in_sizes[1];       // 65536; in_sizes[0]/nrows == NCLS

  float* conf_g = (float*)d_ws;            // 15000 f32
  float* corr_g = conf_g + HBINS;          // 15000 f32

  zero_f32<<<(2 * HBINS + 255) / 256, 256, 0, stream>>>(conf_g, 2 * HBINS);
  cwece_hist<<<HIST_BLOCKS, 256, 0, stream>>>(logits, labels, conf_g, corr_g,
                                              nrows);
  cwece_finalize<<<1, 256, 0, stream>>>(conf_g, corr_g, (float*)d_out, nrows);
}